// Condpf_70300024701596
// MI455X (gfx1250) — compile-verified
//
#include <hip/hip_runtime.h>

#define S_TOT 641
#define NPART 1000
#define DXD   16
#define T_BLK 10
#define SUBST 64
#define HLF   0.015625f   // 2^-6
#define SQHL  0.125f      // sqrt(2^-6)

typedef __attribute__((ext_vector_type(2))) float v2f;
typedef __attribute__((ext_vector_type(8))) float v8f;

// ---------------- counter-based RNG (splitmix64 + Box-Muller) ----------------
__device__ __forceinline__ unsigned long long sm64(unsigned long long z) {
    z += 0x9E3779B97F4A7C15ULL;
    z = (z ^ (z >> 30)) * 0xBF58476D1CE4E5B9ULL;
    z = (z ^ (z >> 27)) * 0x94D049BB133111EBULL;
    return z ^ (z >> 31);
}
__device__ __forceinline__ float u01(unsigned int v) {
    return ((float)(v >> 8) + 0.5f) * (1.0f / 16777216.0f);   // (0,1), never denormal
}
// one hash -> two standard normals (Box-Muller pair), raw HW transcendentals:
//   v_log_f32 is log2(x)  -> ln(u1) = ln2 * log2(u1), folded into the -2 factor
//   v_sin/v_cos take input in revolutions -> feed u2 directly (angle = 2*pi*u2)
__device__ __forceinline__ v2f rnorm2(unsigned long long ctr) {
    unsigned long long z = sm64(ctr);
    float u1 = u01((unsigned int)z);
    float u2 = u01((unsigned int)(z >> 32));
    float r  = __builtin_amdgcn_sqrtf(-1.3862943611f * __builtin_amdgcn_logf(u1));
    v2f out;
    out.x = r * __builtin_amdgcn_cosf(u2);
    out.y = r * __builtin_amdgcn_sinf(u2);
    return out;
}
__device__ __forceinline__ float runif(unsigned long long ctr) {
    return u01((unsigned int)(sm64(ctr) >> 32));
}

// ---------------- Euler-Maruyama segment: Xn = (I + h*A)*X + Sig*DW ----------
// One wave owns a 16(dim) x 16(particle) f32 tile. State chained through a
// per-wave LDS tile; matmuls via V_WMMA_F32_16X16X4_F32 (K=16 -> 4 chunks).
__global__ __launch_bounds__(256) void pf_euler(
    const float* __restrict__ A, const float* __restrict__ Sig,
    const float* __restrict__ input_path, float* __restrict__ un, int t)
{
    __shared__ float xs[8 * 256];
    const int wave = threadIdx.x >> 5;
    const int lane = threadIdx.x & 31;
    const int tile = blockIdx.x * 8 + wave;
    if (tile >= (NPART + 15) / 16) return;          // wave-uniform exit, EXEC stays all-1s
    float* X = xs + wave * 256;

    const int pl   = lane & 15;                     // particle column within tile
    const int hi   = lane >> 4;                     // half-wave select
    const int ksel = hi * 2;
    const int pg   = tile * 16 + pl;                // global particle id (padded to 1008)
    const int start = t * SUBST;

    // load x0 into LDS tile (X[k][p]); padded particles mirror particle N-1
    {
        const int pc = (pg < NPART) ? pg : (NPART - 1);
        const float* s0 = un + ((size_t)start * NPART + pc) * DXD;
        #pragma unroll
        for (int j = 0; j < 8; ++j) { int k = hi * 8 + j; X[k * 16 + pl] = s0[k]; }
    }

    // constant A-matrix operands in WMMA f32 16x4 layout:
    //   v[c].x -> K = 4c+ksel, v[c].y -> K = 4c+1+ksel, M = pl
    v2f Ma[4], Sg[4];
    {
        const int m = pl;
        #pragma unroll
        for (int c = 0; c < 4; ++c) {
            int k0 = 4 * c + ksel, k1 = k0 + 1;
            Ma[c].x = ((m == k0) ? 1.0f : 0.0f) + HLF * A[m * DXD + k0];
            Ma[c].y = ((m == k1) ? 1.0f : 0.0f) + HLF * A[m * DXD + k1];
            Sg[c].x = Sig[m * DXD + k0];
            Sg[c].y = Sig[m * DXD + k1];
        }
    }
    asm volatile("s_wait_dscnt 0x0" ::: "memory");

    for (int st = 0; st < SUBST; ++st) {
        const int gstep = start + st;

        // B-matrix operands: state chunks from LDS + freshly generated noise.
        // RNG counter indexes the (gstep, k-pair, particle) lattice; each hash
        // feeds both elements of a lane's chunk (k0,k1) via the Box-Muller pair.
        v2f Bx[4], Bw[4];
        #pragma unroll
        for (int c = 0; c < 4; ++c) {
            int k0 = 4 * c + ksel, k1 = k0 + 1;
            Bx[c].x = X[k0 * 16 + pl];
            Bx[c].y = X[k1 * 16 + pl];
            v2f nn = rnorm2(((unsigned long long)(gstep * 8 + (k0 >> 1)) << 10) | (unsigned)pg);
            Bw[c].x = SQHL * nn.x;
            Bw[c].y = SQHL * nn.y;
        }

        v8f acc = {};
        #pragma unroll
        for (int c = 0; c < 4; ++c)
            acc = __builtin_amdgcn_wmma_f32_16x16x4_f32(false, Ma[c], false, Bx[c],
                                                        (short)0, acc, false, false);
        #pragma unroll
        for (int c = 0; c < 4; ++c)
            acc = __builtin_amdgcn_wmma_f32_16x16x4_f32(false, Sg[c], false, Bw[c],
                                                        (short)0, acc, false, false);

        // persist step state to global (particle N-1 forced onto the input path)
        const int s = gstep + 1;
        if (pg < NPART) {
            float* dst = un + ((size_t)s * NPART + pg) * DXD + 8 * hi;  // 8 contiguous rows
            if (pg == NPART - 1) {
                const float4* ip = (const float4*)(input_path + (size_t)s * DXD + 8 * hi);
                ((float4*)dst)[0] = ip[0];
                ((float4*)dst)[1] = ip[1];
            } else {
                float4 lo4 = { acc[0], acc[1], acc[2], acc[3] };
                float4 hi4 = { acc[4], acc[5], acc[6], acc[7] };
                ((float4*)dst)[0] = lo4;
                ((float4*)dst)[1] = hi4;
            }
        }

        // D layout (row j / j+8 per half-wave) back into LDS tile for next step
        #pragma unroll
        for (int j = 0; j < 8; ++j) X[(j + 8 * hi) * 16 + pl] = acc[j];
        asm volatile("s_wait_dscnt 0x0" ::: "memory");
    }
}

// -------- weights: gn update, softmax, ESS gate, cumsum bins, multinomial ----
__global__ __launch_bounds__(1024) void pf_weights(
    const float* __restrict__ un, const float* __restrict__ observe_path,
    float* __restrict__ gn, float* __restrict__ bins, int* __restrict__ idx,
    int* __restrict__ flags, int t)
{
    __shared__ float red[1024];
    __shared__ float sc[1024];
    const int n = threadIdx.x;
    const int upd = SUBST * (t + 1);

    float g = -3.0e38f;
    if (n < NPART) {
        const float* x = un + ((size_t)upd * NPART + n) * DXD;
        const float* y = observe_path + (t + 1) * DXD;
        float s = 0.f;
        #pragma unroll
        for (int d = 0; d < DXD; ++d) { float df = y[d] - x[d]; s += df * df; }
        g = gn[n] - 0.5f * s;
    }

    red[n] = g; __syncthreads();
    for (int off = 512; off > 0; off >>= 1) { if (n < off) red[n] = fmaxf(red[n], red[n + off]); __syncthreads(); }
    float mx = red[0]; __syncthreads();

    float e = (n < NPART) ? __expf(g - mx) : 0.f;
    red[n] = e; __syncthreads();
    for (int off = 512; off > 0; off >>= 1) { if (n < off) red[n] += red[n + off]; __syncthreads(); }
    float tot = red[0]; __syncthreads();
    float w = e / tot;

    red[n] = w * w; __syncthreads();
    for (int off = 512; off > 0; off >>= 1) { if (n < off) red[n] += red[n + off]; __syncthreads(); }
    float ssq = red[0]; __syncthreads();
    const int do_rs = (1.0f / ssq <= 0.5f * (float)NPART) ? 1 : 0;

    // inclusive scan (Hillis-Steele) for cumulative weights
    sc[n] = w;
    for (int off = 1; off < 1024; off <<= 1) {
        __syncthreads(); float v = (n >= off) ? sc[n - off] : 0.f;
        __syncthreads(); sc[n] += v;
    }
    __syncthreads();
    if (n == NPART - 1) sc[n] = fmaxf(1.0f, sc[n]);
    __syncthreads();

    if (n < NPART) {
        bins[n] = sc[n];
        float u = runif(0x100000000ULL + (unsigned long long)t * 1024ULL + (unsigned)n);
        int lo = 0, hiB = NPART;                       // searchsorted(side='right')
        while (lo < hiB) { int mid = (lo + hiB) >> 1; if (sc[mid] <= u) lo = mid + 1; else hiB = mid; }
        idx[n] = (lo < NPART) ? lo : NPART - 1;
        gn[n]  = do_rs ? 0.0f : g;
    }
    if (n == 0) flags[0] = do_rs;
}

// -------- resample gather of prefix (or plain copy) into ping-pong buffer ----
__global__ void pf_resample(const float* __restrict__ cur, float* __restrict__ nxt,
                            const float* __restrict__ input_path,
                            const int* __restrict__ idx, const int* __restrict__ flags, int t)
{
    const int upd = SUBST * (t + 1);
    const size_t rows = (size_t)(upd + 1) * NPART;
    size_t r = (size_t)blockIdx.x * blockDim.x + threadIdx.x;
    if (r >= rows) return;
    const int s = (int)(r / NPART);
    const int n = (int)(r % NPART);
    const int do_rs = flags[0];
    float4* dst = (float4*)(nxt + r * DXD);
    const float4* src;
    if (n == NPART - 1) src = (const float4*)(input_path + (size_t)s * DXD);
    else {
        int m = do_rs ? idx[n] : n;
        src = (const float4*)(cur + ((size_t)s * NPART + m) * DXD);
    }
    dst[0] = src[0]; dst[1] = src[1]; dst[2] = src[2]; dst[3] = src[3];
}

// -------- final path selection + gather --------------------------------------
__global__ __launch_bounds__(1024) void pf_final(const float* __restrict__ cur,
                                                 const float* __restrict__ bins,
                                                 float* __restrict__ out)
{
    __shared__ int sh_i0;
    if (threadIdx.x == 0) {
        float u = runif(0x200000000ULL);
        int lo = 0, hiB = NPART;
        while (lo < hiB) { int mid = (lo + hiB) >> 1; if (bins[mid] <= u) lo = mid + 1; else hiB = mid; }
        sh_i0 = (lo < NPART) ? lo : NPART - 1;
    }
    __syncthreads();
    const int i0 = sh_i0;
    for (int e = threadIdx.x; e < S_TOT * DXD; e += 1024) {
        int s = e >> 4, d = e & 15;
        out[e] = cur[((size_t)s * NPART + i0) * DXD + d];
    }
}

__global__ void pf_init(float* __restrict__ u0, float* __restrict__ gn,
                        const float* __restrict__ input_path)
{
    int i = blockIdx.x * blockDim.x + threadIdx.x;
    if (i < NPART * DXD) {
        int n = i >> 4, d = i & 15;
        u0[i] = (n == NPART - 1) ? input_path[d] : 0.0f;   // row s=0
    }
    if (i < NPART) gn[i] = 0.0f;
}

extern "C" void kernel_launch(void* const* d_in, const int* in_sizes, int n_in,
                              void* d_out, int out_size, void* d_ws, size_t ws_size,
                              hipStream_t stream)
{
    const float* input_path   = (const float*)d_in[0];
    const float* observe_path = (const float*)d_in[1];
    const float* A            = (const float*)d_in[2];
    const float* Sig          = (const float*)d_in[3];
    float* out = (float*)d_out;

    const size_t un_bytes = ((size_t)S_TOT * NPART * DXD * sizeof(float) + 255) & ~(size_t)255;
    char* ws = (char*)d_ws;
    float* U0   = (float*)ws;
    float* U1   = (float*)(ws + un_bytes);
    char*  aux  = ws + 2 * un_bytes;
    float* gn   = (float*)aux;
    float* bins = (float*)(aux + 4096);
    int*   idx  = (int*)(aux + 8192);
    int*   flags= (int*)(aux + 12288);

    float* cur = U0;
    float* nxt = U1;

    pf_init<<<(NPART * DXD + 255) / 256, 256, 0, stream>>>(cur, gn, input_path);
    for (int t = 0; t < T_BLK; ++t) {
        pf_euler<<<8, 256, 0, stream>>>(A, Sig, input_path, cur, t);
        pf_weights<<<1, 1024, 0, stream>>>(cur, observe_path, gn, bins, idx, flags, t);
        if (t < T_BLK - 1) {
            const int upd = SUBST * (t + 1);
            const size_t rows = (size_t)(upd + 1) * NPART;
            const int blocks = (int)((rows + 255) / 256);
            pf_resample<<<blocks, 256, 0, stream>>>(cur, nxt, input_path, idx, flags, t);
            float* tmp = cur; cur = nxt; nxt = tmp;
        }
    }
    pf_final<<<1, 1024, 0, stream>>>(cur, bins, out);
}